// S4Layer_30236569763912
// MI455X (gfx1250) — compile-verified
//
#include <hip/hip_runtime.h>
#include <math.h>

// Problem sizes (fixed by the reference)
#define B_  8
#define L_  4096
#define D_  1024
#define N2_ 32

typedef __attribute__((ext_vector_type(2))) float v2f;
typedef __attribute__((ext_vector_type(4))) float v4f;
typedef __attribute__((ext_vector_type(8))) float v8f;
typedef __attribute__((ext_vector_type(4))) int   v4i;

#define AS1 __attribute__((address_space(1)))
#define AS3 __attribute__((address_space(3)))

#if __has_builtin(__builtin_amdgcn_global_load_async_to_lds_b128)
#define HAVE_ASYNC_LD 1
#else
#define HAVE_ASYNC_LD 0
#endif
#if __has_builtin(__builtin_amdgcn_global_store_async_from_lds_b128)
#define HAVE_ASYNC_ST 1
#else
#define HAVE_ASYNC_ST 0
#endif

// 16B global -> LDS (async; tracked by ASYNCcnt)
__device__ __forceinline__ void g2l_b128(void* lds, const void* gaddr) {
#if HAVE_ASYNC_LD
  __builtin_amdgcn_global_load_async_to_lds_b128((AS1 v4i*)gaddr,
                                                 (AS3 v4i*)lds, 0, 0);
#else
  *(v4f*)lds = *(const v4f*)gaddr;
#endif
}

// 16B LDS -> global (async)
__device__ __forceinline__ void l2g_b128(void* gaddr, const void* lds) {
#if HAVE_ASYNC_ST
  __builtin_amdgcn_global_store_async_from_lds_b128((AS1 v4i*)gaddr,
                                                    (AS3 v4i*)lds, 0, 0);
#else
  *(v4f*)gaddr = *(const v4f*)lds;
#endif
}

__device__ __forceinline__ void wait_async0() {
#if HAVE_ASYNC_LD || HAVE_ASYNC_ST
#if __has_builtin(__builtin_amdgcn_s_wait_asynccnt)
  __builtin_amdgcn_s_wait_asynccnt(0);
#else
  asm volatile("s_wait_asynccnt 0x0" ::: "memory");
#endif
#endif
}

// ---------------------------------------------------------------------------
// Kernel 1: per-(d,n) SSM parameters.
//   a  = exp(dt*A);  C~ = (C_real + i C_imag) * (a - 1) / A
// K[t] = 2*Re(sum_n C~_n a_n^t)  ==> conv(x,K) == recurrence h = a*h + x.
// ---------------------------------------------------------------------------
__global__ void s4d_params_kernel(const float* __restrict__ log_dt,
                                  const float* __restrict__ log_A_real,
                                  const float* __restrict__ A_imag,
                                  const float* __restrict__ C_real,
                                  const float* __restrict__ C_imag,
                                  float* __restrict__ par) {
  int id = blockIdx.x * blockDim.x + threadIdx.x;
  if (id >= D_ * N2_) return;
  int d = id / N2_;
  float dt = expf(log_dt[d]);
  float Ar = -expf(log_A_real[id]);
  float Ai = A_imag[id];
  float er = expf(Ar * dt);
  float th = Ai * dt;
  float ar = er * cosf(th);
  float ai = er * sinf(th);
  float nr = ar - 1.0f, ni = ai;
  float inv = 1.0f / (Ar * Ar + Ai * Ai);
  float wr = (nr * Ar + ni * Ai) * inv;
  float wi = (ni * Ar - nr * Ai) * inv;
  float Cr = C_real[id], Ci = C_imag[id];
  par[0 * D_ * N2_ + id] = ar;
  par[1 * D_ * N2_ + id] = ai;
  par[2 * D_ * N2_ + id] = Cr * wr - Ci * wi;
  par[3 * D_ * N2_ + id] = Cr * wi + Ci * wr;
}

// ---------------------------------------------------------------------------
// Kernel 2: SSM scan + exact GELU.
// One wave32 per (b,d) channel; lane n holds complex state of mode n.
// x staged into LDS via async-to-LDS; GELU results pushed back out via
// async-from-LDS so stores overlap the next chunk's x loads.
// ---------------------------------------------------------------------------
#define TC 256
__global__ __launch_bounds__(256) void s4d_scan_kernel(
    const float* __restrict__ x, const float* __restrict__ par,
    float* __restrict__ g) {
  __shared__ __align__(16) float xs[TC * 8];
  __shared__ __align__(16) float ys[TC * 8];

  int b  = blockIdx.x >> 7;            // 128 blocks per batch (D/8)
  int d0 = (blockIdx.x & 127) << 3;    // 8 channels per block
  int tid = threadIdx.x;
  int w  = tid >> 5;                   // wave id -> channel d0+w
  int ln = tid & 31;                   // lane   -> mode n

  int pidx = (d0 + w) * N2_ + ln;
  float ar = par[0 * D_ * N2_ + pidx];
  float ai = par[1 * D_ * N2_ + pidx];
  float cr = par[2 * D_ * N2_ + pidx];
  float ci = par[3 * D_ * N2_ + pidx];
  float hr = 0.0f, hi = 0.0f;

  const float* xb = x + (size_t)b * L_ * D_;
  float*       gb = g + (size_t)b * L_ * D_;

  for (int l0 = 0; l0 < L_; l0 += TC) {
    // stage x[b, l0:l0+TC, d0:d0+8] into LDS (async, 16B per lane)
    for (int idx = tid; idx < TC * 2; idx += 256) {
      int li = idx >> 1, c4 = (idx & 1) << 2;
      g2l_b128(&xs[li * 8 + c4], xb + (size_t)(l0 + li) * D_ + d0 + c4);
    }
    wait_async0();      // xs ready; previous chunk's ys stores drained
    __syncthreads();

    for (int i = 0; i < TC; ++i) {
      float xv = xs[i * 8 + w];               // LDS broadcast within wave
      float tr = ar * hr - ai * hi + xv;      // h = a*h + x  (x real)
      float ti = ar * hi + ai * hr;
      hr = tr; hi = ti;
      float p = cr * hr - ci * hi;            // Re(C~ * h)
      p += __shfl_xor(p, 16, 32);
      p += __shfl_xor(p, 8, 32);
      p += __shfl_xor(p, 4, 32);
      p += __shfl_xor(p, 2, 32);
      p += __shfl_xor(p, 1, 32);
      if (ln == 0) {
        float y = 2.0f * p;
        ys[i * 8 + w] = 0.5f * y * (1.0f + erff(y * 0.70710678118654752f));
      }
    }
    __syncthreads();

    // push g[b, l0:l0+TC, d0:d0+8] from LDS (async; drained next chunk)
    for (int idx = tid; idx < TC * 2; idx += 256) {
      int li = idx >> 1, c4 = (idx & 1) << 2;
      l2g_b128(gb + (size_t)(l0 + li) * D_ + d0 + c4, &ys[li * 8 + c4]);
    }
  }
  wait_async0();
}

// ---------------------------------------------------------------------------
// Kernel 3: out = GELU(y) @ W^T + b  via V_WMMA_F32_16X16X4_F32.
// M=32768, N=1024, K=1024. Block tile 128x128, 8 waves, 8 x v8f acc/wave.
// K chunked by 32 through double-buffered LDS filled with async-to-LDS
// b128 loads (prefetch next chunk while WMMAs consume current one).
// LDS row stride 36 floats = 144B: 16B aligned for b128, and 36*m mod 64
// gives 16 distinct banks for fragment reads.
// Epilogue: transpose tile through LDS, coalesced nontemporal float4 stores.
// ---------------------------------------------------------------------------
#define STRD 36
#define ASZ  (128 * STRD)                  // floats per A/B chunk buffer
#define KC   32
__global__ __launch_bounds__(256) void gemm_wmma_kernel(
    const float* __restrict__ g, const float* __restrict__ Wm,
    const float* __restrict__ bias, float* __restrict__ out) {
  // layout: [As0][Bs0][As1][Bs1]  (4*4608 floats = 72KB); reused for epilogue
  __shared__ __align__(16) float smem[4 * ASZ];

  int mt = blockIdx.x >> 3;   // 256 M-tiles
  int nt = blockIdx.x & 7;    // 8 N-tiles
  int m0 = mt * 128, n0 = nt * 128;
  int tid = threadIdx.x;
  int w  = tid >> 5;
  int ln = tid & 31;
  int lm = ln & 15;           // row/col within 16x16 fragment
  int kk = (ln >> 4) * 2;     // K sub-offset: 0 (lanes 0-15) or 2 (16-31)

  v8f acc[8];
  v8f zero = {0.f, 0.f, 0.f, 0.f, 0.f, 0.f, 0.f, 0.f};
#pragma unroll
  for (int i = 0; i < 8; ++i) acc[i] = zero;

  const float* gA = g  + (size_t)m0 * D_;
  const float* gB = Wm + (size_t)n0 * D_;

  // fill buffer 0 with K-chunk 0
  {
    float* As = smem;            // buf 0
    float* Bs = smem + ASZ;
    for (int idx = tid; idx < 128 * 8; idx += 256) {
      int r = idx >> 3, c4 = (idx & 7) << 2;
      g2l_b128(&As[r * STRD + c4], gA + (size_t)r * D_ + c4);
      g2l_b128(&Bs[r * STRD + c4], gB + (size_t)r * D_ + c4);
    }
    wait_async0();
    __syncthreads();
  }

  const int NCH = D_ / KC;      // 32 chunks
  for (int ch = 0; ch < NCH; ++ch) {
    int cur = ch & 1;
    // prefetch next chunk into the other buffer (overlaps with WMMAs below)
    if (ch + 1 < NCH) {
      int kc = (ch + 1) * KC;
      float* As = smem + ((ch + 1) & 1) * 2 * ASZ;
      float* Bs = As + ASZ;
      for (int idx = tid; idx < 128 * 8; idx += 256) {
        int r = idx >> 3, c4 = (idx & 7) << 2;
        g2l_b128(&As[r * STRD + c4], gA + (size_t)r * D_ + kc + c4);
        g2l_b128(&Bs[r * STRD + c4], gB + (size_t)r * D_ + kc + c4);
      }
    }

    const float* As = smem + cur * 2 * ASZ;
    const float* Bs = As + ASZ;
#pragma unroll
    for (int ks = 0; ks < 8; ++ks) {
      int kb = ks * 4 + kk;
      v2f a;
      a.x = As[(w * 16 + lm) * STRD + kb];
      a.y = As[(w * 16 + lm) * STRD + kb + 1];
#pragma unroll
      for (int ns = 0; ns < 8; ++ns) {
        v2f bf;
        bf.x = Bs[(ns * 16 + lm) * STRD + kb];
        bf.y = Bs[(ns * 16 + lm) * STRD + kb + 1];
        acc[ns] = __builtin_amdgcn_wmma_f32_16x16x4_f32(
            false, a, false, bf, (short)0, acc[ns], false, false);
      }
    }
    wait_async0();      // next chunk landed
    __syncthreads();    // all waves done with cur; safe to reuse
  }

  // epilogue: stage tile (+bias) in LDS, then coalesced NT float4 stores.
  // staging layout: [128 rows][132 floats] = 16896 floats (< 4*ASZ)
  int mhi = (ln >> 4) * 8;
#pragma unroll
  for (int ns = 0; ns < 8; ++ns) {
    float bv = bias[n0 + ns * 16 + lm];
#pragma unroll
    for (int v = 0; v < 8; ++v) {
      smem[(w * 16 + mhi + v) * 132 + ns * 16 + lm] = acc[ns][v] + bv;
    }
  }
  __syncthreads();
  for (int idx = tid; idx < 128 * 32; idx += 256) {
    int r = idx >> 5, c4 = (idx & 31) << 2;
    v4f val = *(const v4f*)&smem[r * 132 + c4];
    __builtin_nontemporal_store(val,
        (v4f*)(out + (size_t)(m0 + r) * D_ + n0 + c4));
  }
}

// ---------------------------------------------------------------------------
// Launch
// ---------------------------------------------------------------------------
extern "C" void kernel_launch(void* const* d_in, const int* in_sizes, int n_in,
                              void* d_out, int out_size, void* d_ws,
                              size_t ws_size, hipStream_t stream) {
  const float* x       = (const float*)d_in[0];
  const float* log_dt  = (const float*)d_in[1];
  const float* log_A   = (const float*)d_in[2];
  const float* A_imag  = (const float*)d_in[3];
  const float* C_real  = (const float*)d_in[4];
  const float* C_imag  = (const float*)d_in[5];
  const float* Wm      = (const float*)d_in[6];
  const float* bias    = (const float*)d_in[7];
  float* out = (float*)d_out;

  // workspace layout: [params: 4*D*N2 floats][g: B*L*D floats]
  float* par = (float*)d_ws;
  float* g   = par + 4 * D_ * N2_;

  s4d_params_kernel<<<(D_ * N2_ + 255) / 256, 256, 0, stream>>>(
      log_dt, log_A, A_imag, C_real, C_imag, par);

  s4d_scan_kernel<<<B_ * (D_ / 8), 256, 0, stream>>>(x, par, g);

  gemm_wmma_kernel<<<(B_ * L_ / 128) * (D_ / 128), 256, 0, stream>>>(
      g, Wm, bias, out);
}